// WideDeep_13451837571106
// MI455X (gfx1250) — compile-verified
//
#include <hip/hip_runtime.h>
#include <hip/hip_bf16.h>
#include <math.h>

typedef __attribute__((ext_vector_type(2))) float v2f;
typedef __attribute__((ext_vector_type(8))) float v8f;

#define NFEAT 100
#define CHUNKS_PER_WAVE 5     // 25 K=4 chunks split over 5 waves
#define WAVES_PER_BLOCK 5
#define THREADS_PER_BLOCK (WAVES_PER_BLOCK * 32)   // 160

// Block = 5 waves, all covering the SAME 16 samples; wave w handles feature
// chunks [5w, 5w+5). Each wave: batch-load 5 index pairs (b64), batch-issue
// 20 z/n gathers, branchless FTRL lazy weights, 5 accumulating
// V_WMMA_F32_16X16X4_F32 (B = ones) for the row-sum. Partials meet in LDS.
__global__ __launch_bounds__(THREADS_PER_BLOCK) void ftrl_wide_forward(
    const int*   __restrict__ Xidx,   // (B, 100) int32 hashed feature ids
    const float* __restrict__ zt,     // (D,) FTRL z state
    const float* __restrict__ nt,     // (D,) FTRL n state
    const float* __restrict__ fc_b,   // (1,) deep-path bias
    float*       __restrict__ out)    // (B, 1) sigmoid output
{
    __shared__ float partial[WAVES_PER_BLOCK * 16];

    const int lane  = threadIdx.x & 31;
    const int wave  = threadIdx.x >> 5;            // 0..4
    const int sbase = blockIdx.x * 16;             // 16 samples per block
    const int m     = lane & 15;                   // A-tile row
    const int khalf = (lane >> 4) << 1;            // lanes 0-15: K+{0,1}; 16-31: K+{2,3}

    // Row base: (sbase+m)*400 bytes -> 16B aligned; +khalf*4 -> 8B aligned.
    const int2* __restrict__ row2 =
        (const int2*)(Xidx + (size_t)(sbase + m) * NFEAT) + (khalf >> 1);

    // ---- stage 1: all index-pair loads in flight ----
    int2 ip[CHUNKS_PER_WAVE];
    #pragma unroll
    for (int i = 0; i < CHUNKS_PER_WAVE; ++i)
        ip[i] = row2[(wave * CHUNKS_PER_WAVE + i) * 2];   // chunk c -> int2 index c*2

    // ---- stage 2: all z/n gathers in flight ----
    float zv[2 * CHUNKS_PER_WAVE], nv[2 * CHUNKS_PER_WAVE];
    #pragma unroll
    for (int i = 0; i < CHUNKS_PER_WAVE; ++i) {
        zv[2 * i]     = zt[ip[i].x];
        zv[2 * i + 1] = zt[ip[i].y];
        nv[2 * i]     = nt[ip[i].x];
        nv[2 * i + 1] = nt[ip[i].y];
    }

    // ---- stage 3: branchless lazy weights + WMMA row-sum ----
    v8f acc = {};
    const v2f bones = {1.0f, 1.0f};
    #pragma unroll
    for (int i = 0; i < CHUNKS_PER_WAVE; ++i) {
        v2f a;
        #pragma unroll
        for (int j = 0; j < 2; ++j) {
            const float zg  = zv[2 * i + j];
            const float ng  = nv[2 * i + j];
            const float sgn = __builtin_copysignf(1.0f, zg);
            // (sign*L1 - z) / ((beta + sqrt(n))/alpha + L2) = (sgn - z)/(10*sqrt(n)+11)
            const float num = sgn - zg;
            const float den = __builtin_amdgcn_sqrtf(ng) * 10.0f + 11.0f;
            const float w   = num * __builtin_amdgcn_rcpf(den);
            a[j] = (__builtin_fabsf(zg) > 1.0f) ? w : 0.0f;
        }
        acc = __builtin_amdgcn_wmma_f32_16x16x4_f32(
                  false, a, false, bones, (short)0, acc, false, false);
    }

    // C/D layout: lane 0 -> rows 0..7 in acc[0..7]; lane 16 -> rows 8..15.
    if ((lane & 15) == 0) {
        const int off = (lane >> 4) * 8;
        #pragma unroll
        for (int r = 0; r < 8; ++r)
            partial[wave * 16 + off + r] = acc[r];
    }
    __syncthreads();

    // ---- finalize: threads 0..15 combine 5 partials, clip + bias + sigmoid ----
    if (threadIdx.x < 16) {
        float s = 0.0f;
        #pragma unroll
        for (int v = 0; v < WAVES_PER_BLOCK; ++v)
            s += partial[v * 16 + threadIdx.x];
        s = fminf(fmaxf(s, -35.0f), 35.0f) + fc_b[0];
        out[sbase + threadIdx.x] = 1.0f / (1.0f + __expf(-s));
    }
}

extern "C" void kernel_launch(void* const* d_in, const int* in_sizes, int n_in,
                              void* d_out, int out_size, void* d_ws, size_t ws_size,
                              hipStream_t stream) {
    const int*   Xidx = (const int*)  d_in[0];   // (B,100) int32
    const float* zt   = (const float*)d_in[1];   // (D,)
    const float* nt   = (const float*)d_in[2];   // (D,)
    // d_in[3] = fc_w (1,1): multiplied by a zero input in the reference -> unused
    const float* fc_b = (const float*)d_in[4];   // (1,)
    float*       out  = (float*)      d_out;     // (B,1)

    const int B = in_sizes[0] / NFEAT;           // 16384
    const int blocks = B / 16;                   // 1024 blocks, 16 samples each
    ftrl_wide_forward<<<blocks, THREADS_PER_BLOCK, 0, stream>>>(Xidx, zt, nt, fc_b, out);
}